// GCN_80470507257933
// MI455X (gfx1250) — compile-verified
//
#include <hip/hip_runtime.h>

// GCN 3-layer forward for MI455X (gfx1250).
// Strategy: GEMM-first reordering (halves layer-0 SpMM traffic), fp32 WMMA
// (v_wmma_f32_16x16x4_f32) for the dense GEMMs, fp32 global atomics for the
// edge scatter (all working buffers fit in the 192MB L2), self-loops folded
// into the finalize pass.

#define NNODES 100000
#define NH     64      // hidden width (all layer outputs)
#define INFEAT 128     // layer-0 input width

typedef float v2f __attribute__((ext_vector_type(2)));
typedef float v8f __attribute__((ext_vector_type(8)));

// ---------------- degree / norm kernels ----------------

__global__ void gcn_init_norms(float* __restrict__ onorm, float* __restrict__ inorm, int n) {
  int i = blockIdx.x * blockDim.x + threadIdx.x;
  if (i < n) { onorm[i] = 1.0f; inorm[i] = 1.0f; }   // self-loop contributes 1 to each degree
}

__global__ void gcn_count_deg(const int* __restrict__ src, const int* __restrict__ dst,
                              float* __restrict__ odeg, float* __restrict__ ideg, int ne) {
  int e = blockIdx.x * blockDim.x + threadIdx.x;
  if (e < ne) {
    atomicAdd(&odeg[src[e]], 1.0f);
    atomicAdd(&ideg[dst[e]], 1.0f);
  }
}

__global__ void gcn_to_norm(float* __restrict__ onorm, float* __restrict__ inorm, int n) {
  int i = blockIdx.x * blockDim.x + threadIdx.x;
  if (i < n) {  // deg >= 1 always (self loop), so max(deg,1) == deg
    onorm[i] = rsqrtf(onorm[i]);
    inorm[i] = rsqrtf(inorm[i]);
  }
}

__global__ void gcn_zero4(float4* __restrict__ p, int n4) {
  int i = blockIdx.x * blockDim.x + threadIdx.x;
  if (i < n4) p[i] = make_float4(0.f, 0.f, 0.f, 0.f);
}

// ---------------- fp32 WMMA GEMM: out[N x 64] = diag(scale) * H[N x K] @ W[K x 64] --------
// grid.x = N/16 row blocks, block = 128 threads = 4 waves; wave w owns 16-col tile w.
// No divergence anywhere: EXEC is all-ones as WMMA requires.
__global__ void gcn_gemm_scaled(const float* __restrict__ H, const float* __restrict__ scale,
                                const float* __restrict__ W, float* __restrict__ out, int K) {
  const int lane = threadIdx.x & 31;
  const int wave = threadIdx.x >> 5;
  const int row0 = blockIdx.x << 4;       // 16-row block
  const int col0 = wave << 4;             // 16-col tile
  const int m    = lane & 15;             // A row / B,C col within tile
  const int hi   = lane >> 4;             // half-wave selector

  const int   row = row0 + m;
  const float s   = scale[row];

  v8f acc = {};                           // C/D 16x16 fp32 accumulator (8 VGPRs)
  const float* hrow = H + (size_t)row * K;

  for (int kb = 0; kb < K; kb += 4) {
    // A(16x4): lane holds A[m][kb + 2*hi + {0,1}] -> contiguous float2, fuse out_norm scale
    float2 a2 = *(const float2*)(hrow + kb + (hi << 1));
    v2f a; a.x = a2.x * s; a.y = a2.y * s;
    // B(4x16): lane holds W[kb + 2*hi + {0,1}][col0 + m]  (W is K x 64 row-major, L2-resident)
    v2f b;
    b.x = W[(kb + (hi << 1) + 0) * NH + col0 + m];
    b.y = W[(kb + (hi << 1) + 1) * NH + col0 + m];
    acc = __builtin_amdgcn_wmma_f32_16x16x4_f32(
        /*neg_a=*/false, a, /*neg_b=*/false, b,
        /*c_mod=*/(short)0, acc, /*reuse_a=*/false, /*reuse_b=*/false);
  }

  // D layout: VGPR r -> row (row0 + r + 8*hi), col (col0 + m)
#pragma unroll
  for (int r = 0; r < 8; ++r) {
    out[(size_t)(row0 + r + (hi << 3)) * NH + col0 + m] = acc[r];
  }
}

// ---------------- SpMM scatter: agg[dst[e]] += tmp[src[e]], 64 fp32 feats/edge ----------
// 16 threads per edge, each moves a float4 (coalesced row reads, L2-resident operands).
__global__ void gcn_spmm(const int* __restrict__ src, const int* __restrict__ dst,
                         const float* __restrict__ tmp, float* __restrict__ agg, int ne) {
  int t  = blockIdx.x * blockDim.x + threadIdx.x;
  int e  = t >> 4;
  if (e >= ne) return;
  int f4 = (t & 15) << 2;
  int s = src[e], d = dst[e];
  const float4 v = *(const float4*)(tmp + (size_t)s * NH + f4);
  float* p = agg + (size_t)d * NH + f4;
  atomicAdd(p + 0, v.x);
  atomicAdd(p + 1, v.y);
  atomicAdd(p + 2, v.z);
  atomicAdd(p + 3, v.w);
}

// ---------------- finalize: h_next = (agg + tmp_selfloop) * in_norm + bias ----------
__global__ void gcn_finalize(const float* __restrict__ agg, const float* __restrict__ tmp,
                             const float* __restrict__ inorm, const float* __restrict__ bias,
                             float* __restrict__ out, int n) {
  int i = blockIdx.x * blockDim.x + threadIdx.x;
  if (i < n) {
    int row = i >> 6, f = i & 63;
    out[i] = (agg[i] + tmp[i]) * inorm[row] + bias[f];
  }
}

__global__ void gcn_copy_ids(const int* __restrict__ ids, int* __restrict__ out, int n) {
  int i = blockIdx.x * blockDim.x + threadIdx.x;
  if (i < n) out[i] = ids[i];
}

// ---------------- orchestration ----------------

extern "C" void kernel_launch(void* const* d_in, const int* in_sizes, int n_in,
                              void* d_out, int out_size, void* d_ws, size_t ws_size,
                              hipStream_t stream) {
  const float* h   = (const float*)d_in[0];
  const int*   src = (const int*)d_in[1];
  const int*   dst = (const int*)d_in[2];
  const int*   ids = (const int*)d_in[3];
  const float* W0  = (const float*)d_in[4];
  const float* b0  = (const float*)d_in[5];
  const float* W1  = (const float*)d_in[6];
  const float* b1  = (const float*)d_in[7];
  const float* W2  = (const float*)d_in[8];
  const float* b2  = (const float*)d_in[9];

  const int N  = NNODES;
  const int ne = in_sizes[1];

  // workspace layout (52.1 MB): norms then two ping-pong N x 64 buffers
  float* onorm = (float*)d_ws;
  float* inorm = onorm + N;
  float* bufA  = inorm + N;                 // 16B-aligned (2*N*4 bytes = 800000*4)
  float* bufB  = bufA + (size_t)N * NH;

  const int nf  = N * NH;                   // 6.4M
  const int nf4 = nf >> 2;
  const int gemmGrid = N / 16;              // 6250, exact
  const int spmmT = ne * 16;

  // degrees -> symmetric norms
  gcn_init_norms<<<(N + 255) / 256, 256, 0, stream>>>(onorm, inorm, N);
  gcn_count_deg<<<(ne + 255) / 256, 256, 0, stream>>>(src, dst, onorm, inorm, ne);
  gcn_to_norm<<<(N + 255) / 256, 256, 0, stream>>>(onorm, inorm, N);

  // ---- layer 0: GEMM-first (128 -> 64) then 64-wide SpMM ----
  gcn_gemm_scaled<<<gemmGrid, 128, 0, stream>>>(h, onorm, W0, bufA, INFEAT);
  gcn_zero4<<<(nf4 + 255) / 256, 256, 0, stream>>>((float4*)bufB, nf4);
  gcn_spmm<<<(spmmT + 255) / 256, 256, 0, stream>>>(src, dst, bufA, bufB, ne);
  gcn_finalize<<<(nf + 255) / 256, 256, 0, stream>>>(bufB, bufA, inorm, b0, bufA, nf); // h1 -> bufA

  // ---- layer 1 ----
  gcn_gemm_scaled<<<gemmGrid, 128, 0, stream>>>(bufA, onorm, W1, bufB, NH);
  gcn_zero4<<<(nf4 + 255) / 256, 256, 0, stream>>>((float4*)bufA, nf4);
  gcn_spmm<<<(spmmT + 255) / 256, 256, 0, stream>>>(src, dst, bufB, bufA, ne);
  gcn_finalize<<<(nf + 255) / 256, 256, 0, stream>>>(bufA, bufB, inorm, b1, bufA, nf); // h2 -> bufA

  // ---- layer 2: finalize straight into d_out ----
  gcn_gemm_scaled<<<gemmGrid, 128, 0, stream>>>(bufA, onorm, W2, bufB, NH);
  gcn_zero4<<<(nf4 + 255) / 256, 256, 0, stream>>>((float4*)bufA, nf4);
  gcn_spmm<<<(spmmT + 255) / 256, 256, 0, stream>>>(src, dst, bufB, bufA, ne);
  gcn_finalize<<<(nf + 255) / 256, 256, 0, stream>>>(bufA, bufB, inorm, b2,
                                                     (float*)d_out, nf);

  // second tuple output: dst_node_ids (raw int bits after the N*64 floats)
  gcn_copy_ids<<<4, 256, 0, stream>>>(ids, (int*)d_out + (size_t)N * NH, 1024);
}